// PatchReader1ConvLayer_20590073217153
// MI455X (gfx1250) — compile-verified
//
#include <hip/hip_runtime.h>

#define N_NODES   50000
#define N_EDGES   800000
#define HIDDEN    128
#define HALFD     64
#define OUT_FEATS 15
#define NUM_GRAPHS 128
#define EPSV      1e-5f
#define SLOPE     0.01f

typedef __attribute__((ext_vector_type(2))) float v2f;
typedef __attribute__((ext_vector_type(8))) float v8f;

__device__ __forceinline__ void atomAddF(float* p, float v) {
    __hip_atomic_fetch_add(p, v, __ATOMIC_RELAXED, __HIP_MEMORY_SCOPE_AGENT);
}
__device__ __forceinline__ float leaky(float x) { return x >= 0.f ? x : SLOPE * x; }

// ---------------- degree computation: deg_out[src]++, deg_in[dst]++ --------
__global__ void __launch_bounds__(256) k_degrees(const int* __restrict__ esrc,
                                                 const int* __restrict__ edst,
                                                 float* __restrict__ deg_out,
                                                 float* __restrict__ deg_in) {
    const int e = blockIdx.x * 256 + threadIdx.x;
    if (e < N_EDGES) {
        atomAddF(deg_out + esrc[e], 1.0f);
        atomAddF(deg_in  + edst[e], 1.0f);
    }
}

// ---------------- h = (X @ Wc) * norm_src  via V_WMMA_F32_16X16X4_F32 ------
// Block = 256 threads = 8 waves. Block owns 16 rows; each wave owns one
// 16x16 N-tile (N=128 -> 8 tiles). 50000 = 3125*16 exactly -> EXEC all ones.
__global__ void __launch_bounds__(256) k_gemm_proj(const float* __restrict__ X,
                                                   const float* __restrict__ Wc,
                                                   const float* __restrict__ deg_out,
                                                   float* __restrict__ H) {
    __shared__ float As[16 * 132];          // padded stride vs 64-bank LDS
    const int tid = threadIdx.x;
    const int m0  = blockIdx.x * 16;
    {   // cooperative A-tile load: 2048 floats, 8 per thread (2x float4)
        const int idx = tid * 8;
        const int r = idx >> 7, c = idx & 127;
        const float4* src = (const float4*)(X + (m0 + r) * HIDDEN + c);
        float4 a0 = src[0], a1 = src[1];
        *(float4*)(&As[r * 132 + c])     = a0;
        *(float4*)(&As[r * 132 + c + 4]) = a1;
    }
    __syncthreads();

    const int wave  = tid >> 5;
    const int lane  = tid & 31;
    const int khalf = lane >> 4;   // 0: K pair {k,k+1}; 1: {k+2,k+3}
    const int ml    = lane & 15;   // A row / B,D column index
    const int ncol  = wave * 16 + ml;

    v8f acc = {};
#pragma unroll
    for (int k = 0; k < HIDDEN; k += 4) {
        const int ka = k + khalf * 2;
        v2f a, b;
        a.x = As[ml * 132 + ka];
        a.y = As[ml * 132 + ka + 1];
        b.x = Wc[(ka)     * HIDDEN + ncol];
        b.y = Wc[(ka + 1) * HIDDEN + ncol];
        acc = __builtin_amdgcn_wmma_f32_16x16x4_f32(
            false, a, false, b, (short)0, acc, false, false);
    }
#pragma unroll
    for (int r = 0; r < 8; ++r) {          // D: vgpr r -> row r + 8*khalf
        const int grow = m0 + r + khalf * 8;
        const float ns = rsqrtf(fmaxf(deg_out[grow], 1.0f));
        H[grow * HIDDEN + ncol] = acc[r] * ns;
    }
}

// ---------------- edge scatter: agg[dst] += h[src] * w  --------------------
// 8 edges per 256-thread block; 32 lanes/edge, 4 feats/lane (float4 gather).
__global__ void __launch_bounds__(256) k_edge_scatter(const float* __restrict__ H,
                                                      const float* __restrict__ ew,
                                                      const int* __restrict__ esrc,
                                                      const int* __restrict__ edst,
                                                      float* __restrict__ agg) {
    const int e    = blockIdx.x * 8 + (threadIdx.x >> 5);
    const int lane = threadIdx.x & 31;
    const int s = esrc[e];
    const int d = edst[e];
    const float w = ew[e];
    const int f0 = lane * 4;
    const float4 hv = *(const float4*)(H + s * HIDDEN + f0);
    float* o = agg + d * HIDDEN + f0;
    atomAddF(o + 0, hv.x * w);
    atomAddF(o + 1, hv.y * w);
    atomAddF(o + 2, hv.z * w);
    atomAddF(o + 3, hv.w * w);
}

// ------ x = leakyrelu(agg*norm_dst) in place + per-feature sum/sumsq -------
__global__ void __launch_bounds__(128) k_act_stats(float* __restrict__ x,
                                                   const float* __restrict__ deg_in,
                                                   float* __restrict__ fsum,
                                                   float* __restrict__ fsumsq) {
    const int f = threadIdx.x;
    const int nbase = blockIdx.x * 64;
    float s = 0.f, ss = 0.f;
    for (int i = 0; i < 64; ++i) {
        const int node = nbase + i;
        if (node < N_NODES) {
            float v = x[node * HIDDEN + f];
            v *= rsqrtf(fmaxf(deg_in[node], 1.0f));
            v = leaky(v);
            x[node * HIDDEN + f] = v;
            s += v; ss += v * v;
        }
    }
    atomAddF(fsum + f, s);
    atomAddF(fsumsq + f, ss);
}

// ---------------- mean / invstd per feature --------------------------------
__global__ void k_finalize_stats(const float* __restrict__ fsum,
                                 const float* __restrict__ fsumsq,
                                 const float* __restrict__ gms,
                                 float* __restrict__ mean,
                                 float* __restrict__ invstd) {
    const int f = threadIdx.x;
    const float m   = fsum[f]   * (1.0f / (float)N_NODES);
    const float ex2 = fsumsq[f] * (1.0f / (float)N_NODES);
    const float sc  = gms[f];
    // E[(x - sc*m)^2] = E[x^2] - 2*sc*m*E[x] + sc^2*m^2
    const float var = ex2 - 2.f * sc * m * m + sc * sc * m * m;
    mean[f]   = m;
    invstd[f] = rsqrtf(var + EPSV);
}

// ---------------- GraphNorm apply + per-graph sum pooling ------------------
__global__ void __launch_bounds__(128) k_pool(const float* __restrict__ x,
                                              const int* __restrict__ gid,
                                              const float* __restrict__ mean,
                                              const float* __restrict__ invstd,
                                              const float* __restrict__ gms,
                                              const float* __restrict__ gnw,
                                              const float* __restrict__ gnb,
                                              float* __restrict__ pooled,
                                              float* __restrict__ counts) {
    const int f = threadIdx.x;
    const int nbase = blockIdx.x * 64;
    const float m = mean[f], is = invstd[f], sc = gms[f], w = gnw[f], b = gnb[f];
    for (int i = 0; i < 64; ++i) {
        const int node = nbase + i;
        if (node < N_NODES) {
            const int g = gid[node];
            const float xn = w * (x[node * HIDDEN + f] - sc * m) * is + b;
            atomAddF(pooled + g * HIDDEN + f, xn);
            if (f == 0) atomAddF(counts + g, 1.0f);
        }
    }
}

// ---------------- head: z = leaky(pooled@Wl^T); out = z@Wcls^T -------------
__global__ void __launch_bounds__(128) k_head(const float* __restrict__ pooled,
                                              const float* __restrict__ counts,
                                              const float* __restrict__ Wl,
                                              const float* __restrict__ Wcls,
                                              float* __restrict__ out) {
    __shared__ float zsh[NUM_GRAPHS * HALFD];
    const int g = threadIdx.x;
    const float inv = 1.0f / fmaxf(counts[g], 1.0f);
    for (int j = 0; j < HALFD; ++j) {
        float acc = 0.f;
        for (int f = 0; f < HIDDEN; ++f)
            acc += pooled[g * HIDDEN + f] * inv * Wl[j * HIDDEN + f];
        zsh[g * HALFD + j] = leaky(acc);
    }
    for (int o = 0; o < OUT_FEATS; ++o) {
        float acc = 0.f;
        for (int j = 0; j < HALFD; ++j)
            acc += zsh[g * HALFD + j] * Wcls[o * HALFD + j];
        out[g * OUT_FEATS + o] = acc;
    }
}

extern "C" void kernel_launch(void* const* d_in, const int* in_sizes, int n_in,
                              void* d_out, int out_size, void* d_ws, size_t ws_size,
                              hipStream_t stream) {
    (void)in_sizes; (void)n_in; (void)out_size; (void)ws_size;
    const float* node_feats = (const float*)d_in[0];
    const float* edge_weight= (const float*)d_in[1];
    const float* Wc         = (const float*)d_in[2];
    const float* gnw        = (const float*)d_in[3];
    const float* gnb        = (const float*)d_in[4];
    const float* gms        = (const float*)d_in[5];
    const float* Wl         = (const float*)d_in[6];
    const float* Wcls       = (const float*)d_in[7];
    const int*   esrc       = (const int*)d_in[8];
    const int*   edst       = (const int*)d_in[9];
    const int*   gid        = (const int*)d_in[10];
    float* out = (float*)d_out;
    float* W   = (float*)d_ws;

    // workspace layout (floats)
    float* deg_out = W;                       // 50000
    float* deg_in  = W + 50000;               // 50000
    float* H       = W + 100000;              // 50000*128
    float* agg     = W + 6500000;             // 50000*128 (becomes x in place)
    float* fsum    = W + 12900000;            // 128
    float* fsumsq  = fsum + 128;              // 128
    float* mean    = fsum + 256;              // 128
    float* invstd  = fsum + 384;              // 128
    float* pooled  = fsum + 512;              // 128*128
    float* counts  = pooled + NUM_GRAPHS * HIDDEN; // 128

    hipMemsetAsync(deg_out, 0, 100000 * sizeof(float), stream);
    hipMemsetAsync(agg, 0, (size_t)N_NODES * HIDDEN * sizeof(float), stream);
    hipMemsetAsync(fsum, 0, (512 + NUM_GRAPHS * HIDDEN + NUM_GRAPHS) * sizeof(float), stream);

    k_degrees     <<<(N_EDGES + 255) / 256, 256, 0, stream>>>(esrc, edst, deg_out, deg_in);
    k_gemm_proj   <<<N_NODES / 16, 256, 0, stream>>>(node_feats, Wc, deg_out, H);
    k_edge_scatter<<<N_EDGES / 8, 256, 0, stream>>>(H, edge_weight, esrc, edst, agg);
    k_act_stats   <<<(N_NODES + 63) / 64, 128, 0, stream>>>(agg, deg_in, fsum, fsumsq);
    k_finalize_stats<<<1, 128, 0, stream>>>(fsum, fsumsq, gms, mean, invstd);
    k_pool        <<<(N_NODES + 63) / 64, 128, 0, stream>>>(agg, gid, mean, invstd, gms, gnw, gnb, pooled, counts);
    k_head        <<<1, 128, 0, stream>>>(pooled, counts, Wl, Wcls, out);
}